// GCN_5222680232280
// MI455X (gfx1250) — compile-verified
//
#include <hip/hip_runtime.h>
#include <hip/hip_bf16.h>

#define NUM_GRAPHS 128
#define HDIM 64

typedef float v2f __attribute__((ext_vector_type(2)));
typedef float v8f __attribute__((ext_vector_type(8)));

// ---------------------------------------------------------------------------
// Degree accumulation: deg[col[e]] += 1  (self-loops folded in later as +1)
// ---------------------------------------------------------------------------
__global__ void deg_kernel(const int* __restrict__ col, float* __restrict__ deg, int E) {
    int i = blockIdx.x * blockDim.x + threadIdx.x;
    int stride = gridDim.x * blockDim.x;
    for (int e = i; e < E; e += stride) {
        atomicAdd(&deg[col[e]], 1.0f);
    }
}

// dinv[i] = rsqrt(deg[i] + 1)   (+1 = self loop; always > 0)
__global__ void rsqrt_kernel(float* __restrict__ dinv, int N) {
    int i = blockIdx.x * blockDim.x + threadIdx.x;
    if (i < N) dinv[i] = rsqrtf(dinv[i] + 1.0f);
}

// ---------------------------------------------------------------------------
// WMMA GEMM: Y[N x 64] = X[N x K] * W[K x 64], fp32, V_WMMA_F32_16X16X4_F32.
// Block = 128 threads (4 waves). Each wave owns one 16-wide column tile of
// the 16 x 64 output slab for this block. W is staged in LDS.
// ---------------------------------------------------------------------------
__global__ void gemm_wmma_f32(const float* __restrict__ X,
                              const float* __restrict__ W,
                              float* __restrict__ Y,
                              int N, int K) {
    __shared__ float sW[128 * HDIM];  // up to K=128: 32 KB
    const int tid = threadIdx.x;
    // cooperative load of W into LDS
    for (int i = tid; i < K * HDIM; i += 128) sW[i] = W[i];
    __syncthreads();

    const int wave = tid >> 5;          // 0..3 -> output column tile
    const int lane = tid & 31;
    const int hf   = lane >> 4;         // lane half: 0 or 1
    const int mr   = lane & 15;         // row within A half
    const int nc   = lane & 15;         // col within B/C tile
    const int m0   = blockIdx.x * 16;
    const int n0   = wave * 16;

    // clamped load row (stores are guarded; keeps EXEC uniform for WMMA)
    int rload = m0 + mr;
    if (rload >= N) rload = N - 1;
    const float* xrow = X + (size_t)rload * K;

    v8f acc = {};
    for (int k = 0; k < K; k += 4) {
        // A 16x4 f32: VGPR0 = K0 (lanes 0-15) / K2 (lanes 16-31), VGPR1 = K1/K3
        v2f a;
        const float2 av = *reinterpret_cast<const float2*>(xrow + k + 2 * hf);
        a.x = av.x;
        a.y = av.y;
        // B 4x16 f32: same K striping across lane halves, N across lanes
        v2f b;
        b.x = sW[(k + 2 * hf) * HDIM + n0 + nc];
        b.y = sW[(k + 2 * hf + 1) * HDIM + n0 + nc];
        acc = __builtin_amdgcn_wmma_f32_16x16x4_f32(
            /*neg_a=*/false, a, /*neg_b=*/false, b,
            /*c_mod=*/(short)0, acc, /*reuse_a=*/false, /*reuse_b=*/false);
    }

    // C/D 16x16 f32: VGPR j holds M=j (lanes 0-15) and M=j+8 (lanes 16-31)
    for (int j = 0; j < 8; ++j) {
        int rstore = m0 + j + 8 * hf;
        if (rstore < N) Y[(size_t)rstore * HDIM + n0 + nc] = acc[j];
    }
}

// ---------------------------------------------------------------------------
// Edge aggregation: agg[col[e]] += h[row[e]] * dinv[row[e]] * dinv[col[e]]
// One wave per edge; lane owns a float2 of the 64-feature payload.
// h rows live in L2 (25.6 MB << 192 MB); atomics resolve in L2.
// ---------------------------------------------------------------------------
__global__ void edge_agg_kernel(const int* __restrict__ row,
                                const int* __restrict__ col,
                                const float* __restrict__ dinv,
                                const float* __restrict__ h,
                                float* __restrict__ agg, int E) {
    const int wid   = (blockIdx.x * blockDim.x + threadIdx.x) >> 5;
    const int lane  = threadIdx.x & 31;
    const int nwave = (gridDim.x * blockDim.x) >> 5;
    for (int e = wid; e < E; e += nwave) {
        const int r = row[e];
        const int c = col[e];
        // prefetch the next gathered row this wave will touch (global_prefetch_b8)
        int en = e + nwave;
        if (en < E) __builtin_prefetch(h + (size_t)row[en] * HDIM, 0, 0);
        const float w = dinv[r] * dinv[c];
        const float2 v = reinterpret_cast<const float2*>(h + (size_t)r * HDIM)[lane];
        float* dst = agg + (size_t)c * HDIM + 2 * lane;
        atomicAdd(dst + 0, v.x * w);
        atomicAdd(dst + 1, v.y * w);
    }
}

// out = relu(agg + h * dinv^2 + bias)   (self-loop term folded in, no atomics)
__global__ void selfloop_bias_relu(const float* __restrict__ agg,
                                   const float* __restrict__ h,
                                   const float* __restrict__ dinv,
                                   const float* __restrict__ bias,
                                   float* __restrict__ out, int N) {
    size_t idx = (size_t)blockIdx.x * blockDim.x + threadIdx.x;
    size_t total = (size_t)N * HDIM;
    size_t stride = (size_t)gridDim.x * blockDim.x;
    for (; idx < total; idx += stride) {
        int i = (int)(idx >> 6);
        int f = (int)(idx & 63);
        float di = dinv[i];
        float v = agg[idx] + h[idx] * di * di + bias[f];
        out[idx] = v > 0.0f ? v : 0.0f;
    }
}

// ---------------------------------------------------------------------------
// Global mean pool: sums + counts per graph (batch is sorted, dense atomics
// into 128x64 live in L2).
// ---------------------------------------------------------------------------
__global__ void pool_kernel(const float* __restrict__ h,
                            const int* __restrict__ batch,
                            float* __restrict__ pool,
                            float* __restrict__ cnt, int N) {
    size_t idx = (size_t)blockIdx.x * blockDim.x + threadIdx.x;
    size_t total = (size_t)N * HDIM;
    size_t stride = (size_t)gridDim.x * blockDim.x;
    for (; idx < total; idx += stride) {
        int i = (int)(idx >> 6);
        int f = (int)(idx & 63);
        int g = batch[i];
        atomicAdd(&pool[g * HDIM + f], h[idx]);
        if (f == 0) atomicAdd(&cnt[g], 1.0f);
    }
}

// logits = (pool/cnt) @ Wl + bl ; softmax. One thread per graph.
__global__ void classify_kernel(const float* __restrict__ pool,
                                const float* __restrict__ cnt,
                                const float* __restrict__ Wl,
                                const float* __restrict__ bl,
                                float* __restrict__ out) {
    int g = threadIdx.x;
    if (g >= NUM_GRAPHS) return;
    float inv = 1.0f / fmaxf(cnt[g], 1.0f);
    float p[HDIM];
    for (int k = 0; k < HDIM; ++k) p[k] = pool[g * HDIM + k] * inv;
    float logits[10];
    float mx = -1e30f;
    for (int c = 0; c < 10; ++c) {
        float s = bl[c];
        for (int k = 0; k < HDIM; ++k) s += p[k] * Wl[k * 10 + c];
        logits[c] = s;
        mx = fmaxf(mx, s);
    }
    float sum = 0.0f;
    for (int c = 0; c < 10; ++c) {
        logits[c] = __expf(logits[c] - mx);
        sum += logits[c];
    }
    float rs = 1.0f / sum;
    for (int c = 0; c < 10; ++c) out[g * 10 + c] = logits[c] * rs;
}

// ---------------------------------------------------------------------------
extern "C" void kernel_launch(void* const* d_in, const int* in_sizes, int n_in,
                              void* d_out, int out_size, void* d_ws, size_t ws_size,
                              hipStream_t stream) {
    const float* x    = (const float*)d_in[0];
    const int*   ei   = (const int*)d_in[1];
    const int*   batch = (const int*)d_in[2];
    const float* W0   = (const float*)d_in[3];
    const float* b0   = (const float*)d_in[4];
    const float* W1   = (const float*)d_in[5];
    const float* b1   = (const float*)d_in[6];
    const float* Wl   = (const float*)d_in[7];
    const float* bl   = (const float*)d_in[8];
    float* out = (float*)d_out;

    const int N = in_sizes[0] / 128;   // 100000
    const int E = in_sizes[1] / 2;     // 3200000
    const int F = 128;

    const int* row = ei;
    const int* col = ei + E;

    // workspace layout (floats)
    float* ws     = (float*)d_ws;
    float* dinv   = ws;                                  // N
    float* bufA   = dinv + N;                            // N*64  (GEMM out)
    float* bufAgg = bufA + (size_t)N * HDIM;             // N*64  (scatter acc)
    float* bufB   = bufAgg + (size_t)N * HDIM;           // N*64  (relu out)
    float* pool   = bufB + (size_t)N * HDIM;             // 128*64
    float* cnt    = pool + NUM_GRAPHS * HDIM;            // 128

    const size_t rowBytes = (size_t)N * HDIM * sizeof(float);

    // ---- degrees -> dinv
    hipMemsetAsync(dinv, 0, (size_t)N * sizeof(float), stream);
    deg_kernel<<<4096, 256, 0, stream>>>(col, dinv, E);
    rsqrt_kernel<<<(N + 255) / 256, 256, 0, stream>>>(dinv, N);

    // ---- layer 1: bufA = x @ W0 (K=128)
    gemm_wmma_f32<<<(N + 15) / 16, 128, 0, stream>>>(x, W0, bufA, N, F);
    hipMemsetAsync(bufAgg, 0, rowBytes, stream);
    edge_agg_kernel<<<8192, 256, 0, stream>>>(row, col, dinv, bufA, bufAgg, E);
    selfloop_bias_relu<<<8192, 256, 0, stream>>>(bufAgg, bufA, dinv, b0, bufB, N);

    // ---- layer 2: bufA = bufB @ W1 (K=64)
    gemm_wmma_f32<<<(N + 15) / 16, 128, 0, stream>>>(bufB, W1, bufA, N, HDIM);
    hipMemsetAsync(bufAgg, 0, rowBytes, stream);
    edge_agg_kernel<<<8192, 256, 0, stream>>>(row, col, dinv, bufA, bufAgg, E);
    selfloop_bias_relu<<<8192, 256, 0, stream>>>(bufAgg, bufA, dinv, b1, bufB, N);

    // ---- mean pool + classifier + softmax
    hipMemsetAsync(pool, 0, (size_t)(NUM_GRAPHS * HDIM + NUM_GRAPHS) * sizeof(float), stream);
    pool_kernel<<<8192, 256, 0, stream>>>(bufB, batch, pool, cnt, N);
    classify_kernel<<<1, NUM_GRAPHS, 0, stream>>>(pool, cnt, Wl, bl, out);
}